// DrCounter_14594298871946
// MI455X (gfx1250) — compile-verified
//
#include <hip/hip_runtime.h>

typedef float v2f __attribute__((ext_vector_type(2)));
typedef float v8f __attribute__((ext_vector_type(8)));

#define WEIGHT_F 200.0f

// ---------------------------------------------------------------------------
// Workspace layout (floats):
//   [0    , 512 ) : accAbs[scale][patch]   (4 x 128)
//   [512  , 1024) : accCnt[scale][patch]
//   [1024 , 1536) : accSq [scale][patch]
//   [1536 , 1664) : mask  (int[128])
// ---------------------------------------------------------------------------

__global__ void init_ws_kernel(float* __restrict__ w, int n) {
    int i = blockIdx.x * blockDim.x + threadIdx.x;
    if (i < n) w[i] = 0.0f;
}

// ---------------------------------------------------------------------------
// Pass A: per (b, patch) reduce |p-g|*pos, pos, (g-p)^2 over the patch.
// Reduction is performed by V_WMMA_F32_16X16X4_F32 against a ones B-matrix.
// Each lane feeds 2 f32 values per WMMA (the 16x4 f32 A operand = 2 VGPRs).
// EXEC is all-ones throughout (uniform trip counts, no tails).
// ---------------------------------------------------------------------------
__launch_bounds__(256)
__global__ void patch_stats_kernel(const float* __restrict__ pre,
                                   const float* __restrict__ gt,
                                   float* __restrict__ accAbs,
                                   float* __restrict__ accCnt,
                                   float* __restrict__ accSq,
                                   int w,            // image width == height
                                   int kh,           // patch height
                                   int kwLog2,       // log2(patch width)
                                   int itersPerSlice)
{
    const int tid   = threadIdx.x;          // 0..255
    const int p     = blockIdx.x;           // patch id: b*8 + l  (0..127)
    const int slice = blockIdx.y;

    const int b  = p >> 3;
    const int l  = p & 7;
    const int py = l >> 2;                  // ph = 2
    const int px = l & 3;                   // pw = 4
    const int kw = 1 << kwLog2;

    const size_t base = (size_t)b * (size_t)w * (size_t)w
                      + (size_t)(py * kh) * (size_t)w
                      + (size_t)(px * kw);
    const float* __restrict__ preB = pre + base;
    const float* __restrict__ gtB  = gt  + base;

    v8f cAbs = {0.f,0.f,0.f,0.f,0.f,0.f,0.f,0.f};
    v8f cCnt = {0.f,0.f,0.f,0.f,0.f,0.f,0.f,0.f};
    v8f cSq  = {0.f,0.f,0.f,0.f,0.f,0.f,0.f,0.f};
    v2f ones = {1.0f, 1.0f};

    int e = (slice * itersPerSlice) * 512 + tid * 2;   // linear index in patch
    for (int it = 0; it < itersPerSlice; ++it, e += 512) {
        const int r = e >> kwLog2;
        const int c = e & (kw - 1);
        const v2f pv  = *(const v2f*)(preB + (size_t)r * w + c);
        const v2f gv0 = *(const v2f*)(gtB  + (size_t)r * w + c);

        v2f aAbs, aCnt, aSq;
#pragma unroll
        for (int k = 0; k < 2; ++k) {
            const float g   = gv0[k] * WEIGHT_F;
            const float d   = pv[k] - g;
            const float pos = (g > 0.0f) ? 1.0f : 0.0f;
            aAbs[k] = fabsf(d) * pos;
            aCnt[k] = pos;
            aSq[k]  = d * d;
        }
        cAbs = __builtin_amdgcn_wmma_f32_16x16x4_f32(false, aAbs, false, ones,
                                                     (short)0, cAbs, false, false);
        cCnt = __builtin_amdgcn_wmma_f32_16x16x4_f32(false, aCnt, false, ones,
                                                     (short)0, cCnt, false, false);
        cSq  = __builtin_amdgcn_wmma_f32_16x16x4_f32(false, aSq,  false, ones,
                                                     (short)0, cSq,  false, false);
    }

    // D columns are identical; lanes 0-15 hold rows 0-7, lanes 16-31 rows 8-15.
    float sA = 0.f, sC = 0.f, sS = 0.f;
#pragma unroll
    for (int k = 0; k < 8; ++k) { sA += cAbs[k]; sC += cCnt[k]; sS += cSq[k]; }

    __shared__ float shA[256], shC[256], shS[256];
    shA[tid] = sA; shC[tid] = sC; shS[tid] = sS;
    __syncthreads();

    if (tid == 0) {
        float tA = 0.f, tC = 0.f, tS = 0.f;
        for (int wv = 0; wv < 256; wv += 32) {
            tA += shA[wv] + shA[wv + 16];
            tC += shC[wv] + shC[wv + 16];
            tS += shS[wv] + shS[wv + 16];
        }
        atomicAdd(&accAbs[p], tA);
        atomicAdd(&accCnt[p], tC);
        atomicAdd(&accSq[p],  tS);
    }
}

// ---------------------------------------------------------------------------
// Pass B: per-patch argmin over scales (first-index tie-break) + scalar loss.
// One block, 128 threads (one per (b, patch)).
// ---------------------------------------------------------------------------
__global__ void route_and_loss_kernel(const float* __restrict__ ws,
                                      int* __restrict__ mask,
                                      float* __restrict__ lossOut)
{
    const int t = threadIdx.x;   // 0..127  == b*8 + l
    __shared__ float selSq[4], selCnt[4];
    if (t < 4) { selSq[t] = 0.0f; selCnt[t] = 0.0f; }
    __syncthreads();

    float best = 3.4e38f;
    int   bi   = 0;
    float sq[4];
    for (int i = 0; i < 4; ++i) {
        const float a = ws[i * 128 + t];
        const float c = ws[512 + i * 128 + t];
        sq[i] = ws[1024 + i * 128 + t];
        const float e = a / (c + 0.1f);
        if (e < best) { best = e; bi = i; }
    }
    mask[t] = bi;
    atomicAdd(&selSq[bi],  sq[bi]);
    atomicAdd(&selCnt[bi], 1.0f);
    __syncthreads();

    if (t == 0) {
        float loss = 0.0f;
        float wgt  = 0.5f;
        for (int i = 0; i < 4; ++i) {
            const float patchElems = (float)((512 >> i) * (256 >> i));
            loss += wgt * selSq[i] / (selCnt[i] * patchElems + 0.01f);
            wgt *= 0.5f;
        }
        lossOut[0] = loss;
    }
}

// ---------------------------------------------------------------------------
// Pass C: assemble out_img and lab_img (B,1,1024,1024 each), /WEIGHT,
// pad value 0.2/WEIGHT = 0.001 outside the centered inner region for s>0.
// float4 per thread; pad boundaries (64/96/112) are 4-aligned so a group is
// uniformly inside or outside.
// ---------------------------------------------------------------------------
__launch_bounds__(256)
__global__ void assemble_kernel(const float* __restrict__ pre0, const float* __restrict__ pre1,
                                const float* __restrict__ pre2, const float* __restrict__ pre3,
                                const float* __restrict__ gt0,  const float* __restrict__ gt1,
                                const float* __restrict__ gt2,  const float* __restrict__ gt3,
                                const int* __restrict__ mask,
                                float* __restrict__ outImg,
                                float* __restrict__ labImg)
{
    const long long g4  = (long long)blockIdx.x * 256 + threadIdx.x;
    const long long pix = g4 * 4;                       // first pixel of group
    const int b   = (int)(pix >> 20);                   // 1024*1024 per image
    const int rem = (int)(pix & ((1 << 20) - 1));
    const int y   = rem >> 10;
    const int x   = rem & 1023;
    const int py  = y >> 9;                             // /512
    const int px  = x >> 8;                             // /256
    const int s   = mask[b * 8 + py * 4 + px];

    float4 o, lb;
    if (s == 0) {
        const float4 pv = *(const float4*)(pre0 + pix);
        const float4 gv = *(const float4*)(gt0  + pix);
        o.x = pv.x / WEIGHT_F; o.y = pv.y / WEIGHT_F;
        o.z = pv.z / WEIGHT_F; o.w = pv.w / WEIGHT_F;
        lb = gv;                                        // gt*200/200
    } else {
        const int kh   = 512 >> s, kw = 256 >> s;
        const int padH = (512 - kh) >> 1, padW = (256 - kw) >> 1;
        const int r = y - (py << 9);
        const int c = x - (px << 8);
        const bool inside = (r >= padH) && (r < padH + kh) &&
                            (c >= padW) && (c < padW + kw);
        if (inside) {
            const float* pp; const float* gg;
            if (s == 1)      { pp = pre1; gg = gt1; }
            else if (s == 2) { pp = pre2; gg = gt2; }
            else             { pp = pre3; gg = gt3; }
            const int wS = 1024 >> s;
            const long long off = (long long)b * wS * wS
                                + (long long)(py * kh + (r - padH)) * wS
                                + (long long)(px * kw + (c - padW));
            const float4 pv = *(const float4*)(pp + off);
            const float4 gv = *(const float4*)(gg + off);
            o.x = pv.x / WEIGHT_F; o.y = pv.y / WEIGHT_F;
            o.z = pv.z / WEIGHT_F; o.w = pv.w / WEIGHT_F;
            lb = gv;
        } else {
            o.x = o.y = o.z = o.w = 0.001f;             // 0.2 / 200
            lb.x = lb.y = lb.z = lb.w = 0.001f;
        }
    }
    *(float4*)(outImg + pix) = o;
    *(float4*)(labImg + pix) = lb;
}

// ---------------------------------------------------------------------------
extern "C" void kernel_launch(void* const* d_in, const int* in_sizes, int n_in,
                              void* d_out, int out_size, void* d_ws, size_t ws_size,
                              hipStream_t stream) {
    // setup_inputs() dict order is interleaved: pre0, gt0, pre1, gt1, ...
    const float* pre[4] = { (const float*)d_in[0], (const float*)d_in[2],
                            (const float*)d_in[4], (const float*)d_in[6] };
    const float* gt[4]  = { (const float*)d_in[1], (const float*)d_in[3],
                            (const float*)d_in[5], (const float*)d_in[7] };

    float* W    = (float*)d_ws;
    int*   mask = (int*)(W + 1536);
    float* outF = (float*)d_out;
    float* lossP   = outF;                         // [0]
    float* outImg  = outF + 1;                     // 16*1024*1024
    float* labImg  = outF + 1 + (1 << 24);         // 16*1024*1024

    init_ws_kernel<<<6, 256, 0, stream>>>(W, 1536);

    const int   slices[4] = {32, 16, 8, 4};
    const int   iters[4]  = {8, 4, 2, 1};          // patchElems / (512*slices)
    for (int i = 0; i < 4; ++i) {
        const int wImg   = 1024 >> i;
        const int kh     = 512 >> i;
        const int kwLog2 = 8 - i;
        patch_stats_kernel<<<dim3(128, slices[i]), 256, 0, stream>>>(
            pre[i], gt[i],
            W + i * 128, W + 512 + i * 128, W + 1024 + i * 128,
            wImg, kh, kwLog2, iters[i]);
    }

    route_and_loss_kernel<<<1, 128, 0, stream>>>(W, mask, lossP);

    // 16 * 1024 * 1024 pixels / 4 per thread / 256 threads = 16384 blocks
    assemble_kernel<<<16384, 256, 0, stream>>>(
        pre[0], pre[1], pre[2], pre[3],
        gt[0],  gt[1],  gt[2],  gt[3],
        mask, outImg, labImg);
}